// LightGCN_49383533970015
// MI455X (gfx1250) — compile-verified
//
#include <hip/hip_runtime.h>

#define NUM_USERS 100000
#define NUM_ITEMS 200000
#define EMB_DIM   64
#define N_TOTAL   (NUM_USERS + NUM_ITEMS)
#define NNZ_TOTAL 10000000
#define BATCH     8192

typedef __attribute__((ext_vector_type(2))) float v2f;
typedef __attribute__((ext_vector_type(8))) float v8f;

// ---------------------------------------------------------------------------
// init: x_cur = acc = concat(user_emb, item_emb); x_next = 0
// float4-vectorized: N_TOTAL*64/4 = 4.8M work-items
// ---------------------------------------------------------------------------
__global__ void lgcn_init(const float* __restrict__ ue, const float* __restrict__ ie,
                          float* __restrict__ xcur, float* __restrict__ acc,
                          float* __restrict__ xnext) {
    long i = (long)blockIdx.x * blockDim.x + threadIdx.x;
    const long n4 = (long)N_TOTAL * EMB_DIM / 4;
    if (i >= n4) return;
    const long u4 = (long)NUM_USERS * EMB_DIM / 4;
    float4 v = (i < u4) ? ((const float4*)ue)[i] : ((const float4*)ie)[i - u4];
    ((float4*)xcur)[i]  = v;
    ((float4*)acc)[i]   = v;
    ((float4*)xnext)[i] = make_float4(0.f, 0.f, 0.f, 0.f);
}

// ---------------------------------------------------------------------------
// SpMM scatter: 16 lanes per edge, float4 per lane (64 dims).
// Edge scalars coalesce (identical address within lane group); the 256B row
// gather and the atomic scatter are L2-resident (x buffers fit in 192MB L2).
// ---------------------------------------------------------------------------
__global__ void lgcn_spmm(const float* __restrict__ vals, const int* __restrict__ rows,
                          const int* __restrict__ cols, const float* __restrict__ x,
                          float* __restrict__ y, int nnz) {
    int t    = blockIdx.x * blockDim.x + threadIdx.x;
    int e    = t >> 4;           // 16 lanes per edge
    int lane = t & 15;           // dims [lane*4, lane*4+3]
    if (e >= nnz) return;

    // Stream-ahead hint for the three edge arrays (global_prefetch_b8).
    if ((threadIdx.x & 63) == 0) {
        int ep = e + 8192; if (ep >= nnz) ep = nnz - 1;
        __builtin_prefetch(vals + ep, 0, 0);
        __builtin_prefetch(rows + ep, 0, 0);
        __builtin_prefetch(cols + ep, 0, 0);
    }

    float v = vals[e];
    int   r = rows[e];
    int   c = cols[e];

    float4 m = ((const float4*)(x + (long)c * EMB_DIM))[lane];
    float* yb = y + (long)r * EMB_DIM + lane * 4;
    atomicAdd(yb + 0, v * m.x);
    atomicAdd(yb + 1, v * m.y);
    atomicAdd(yb + 2, v * m.z);
    atomicAdd(yb + 3, v * m.w);
}

// ---------------------------------------------------------------------------
// acc += x_next; zero the retiring buffer (becomes next layer's scatter target)
// ---------------------------------------------------------------------------
__global__ void lgcn_accum_zero(float* __restrict__ acc, const float* __restrict__ xnext,
                                float* __restrict__ oldcur, long n4) {
    long i = (long)blockIdx.x * blockDim.x + threadIdx.x;
    if (i >= n4) return;
    float4 a = ((float4*)acc)[i];
    float4 b = ((const float4*)xnext)[i];
    a.x += b.x; a.y += b.y; a.z += b.z; a.w += b.w;
    ((float4*)acc)[i]    = a;
    ((float4*)oldcur)[i] = make_float4(0.f, 0.f, 0.f, 0.f);
}

// ---------------------------------------------------------------------------
// Scoring via V_WMMA_F32_16X16X4_F32 (exact f32 math):
// one wave handles 16 (u,i) pairs; D = U_tile(16x64) x I_tile^T(64x16)
// accumulated over 16 K=4 steps; diagonal = the 16 wanted dot products.
//
// A 16x4 layout : lanes 0-15 -> M=lane, VGPR{0,1}=K{0,1}; lanes 16-31 K{2,3}
// B 4x16 layout : lanes 0-15 -> N=lane, VGPR{0,1}=K{0,1}; lanes 16-31 K{2,3}
//   => both fragments load float2 at row[4*kk + (lane<16 ? 0 : 2)]
// C/D layout    : VGPR r, lanes 0-15: (M=r,N=lane); lanes 16-31: (M=r+8,N=lane-16)
//   => diag in lanes 0-7 at c[lane], lanes 24-31 at c[lane-24]
// Reference scales out by 1/4 before the dot => score = dot/16.
// ---------------------------------------------------------------------------
__global__ void lgcn_score(const float* __restrict__ acc, const int* __restrict__ u_idx,
                           const int* __restrict__ i_idx, float* __restrict__ out) {
    int tid  = threadIdx.x;
    int wave = tid >> 5;
    int lane = tid & 31;
    int base = (blockIdx.x * (blockDim.x >> 5) + wave) * 16;

    int m  = lane & 15;
    int ko = (lane < 16) ? 0 : 2;
    const float* uvec = acc + (long)u_idx[base + m] * EMB_DIM;
    const float* ivec = acc + ((long)NUM_USERS + i_idx[base + m]) * EMB_DIM;

    v8f c = {0.f, 0.f, 0.f, 0.f, 0.f, 0.f, 0.f, 0.f};
#pragma unroll
    for (int kk = 0; kk < 16; ++kk) {
        v2f a = *(const v2f*)(uvec + 4 * kk + ko);
        v2f b = *(const v2f*)(ivec + 4 * kk + ko);
        c = __builtin_amdgcn_wmma_f32_16x16x4_f32(
                /*neg_a=*/false, a, /*neg_b=*/false, b,
                /*c_mod=*/(short)0, c, /*reuse_a=*/false, /*reuse_b=*/false);
    }

    float cl[8];
#pragma unroll
    for (int r = 0; r < 8; ++r) cl[r] = c[r];

    if (lane < 8)
        out[base + lane] = cl[lane] * 0.0625f;
    else if (lane >= 24)
        out[base + (lane - 16)] = cl[lane - 24] * 0.0625f;
}

// ---------------------------------------------------------------------------
extern "C" void kernel_launch(void* const* d_in, const int* in_sizes, int n_in,
                              void* d_out, int out_size, void* d_ws, size_t ws_size,
                              hipStream_t stream) {
    const float* ue   = (const float*)d_in[0];
    const float* ie   = (const float*)d_in[1];
    const float* vals = (const float*)d_in[2];
    const int*   rows = (const int*)d_in[3];
    const int*   cols = (const int*)d_in[4];
    const int*   uix  = (const int*)d_in[5];
    const int*   iix  = (const int*)d_in[6];
    float*       out  = (float*)d_out;

    const size_t bufElems = (size_t)N_TOTAL * EMB_DIM;   // 19.2M floats = 76.8MB
    float* bufA = (float*)d_ws;
    float* bufB = bufA + bufElems;
    float* acc  = bufB + bufElems;

    float* xcur  = bufA;
    float* xnext = bufB;

    const long n4 = (long)bufElems / 4;
    const int  vb = (int)((n4 + 255) / 256);

    lgcn_init<<<vb, 256, 0, stream>>>(ue, ie, xcur, acc, xnext);

    const int spmm_blocks = (int)(((long)NNZ_TOTAL * 16 + 255) / 256);
    for (int layer = 0; layer < 3; ++layer) {
        lgcn_spmm<<<spmm_blocks, 256, 0, stream>>>(vals, rows, cols, xcur, xnext, NNZ_TOTAL);
        lgcn_accum_zero<<<vb, 256, 0, stream>>>(acc, xnext, xcur, n4);
        float* t = xcur; xcur = xnext; xnext = t;   // retiring buffer is now zeroed
    }

    lgcn_score<<<BATCH / 128, 256, 0, stream>>>(acc, uix, iix, out);
}